// SoftDTW_10977936408749
// MI455X (gfx1250) — compile-verified
//
#include <hip/hip_runtime.h>
#include <hip/hip_bf16.h>
#include <math.h>

typedef __attribute__((ext_vector_type(2))) float v2f;
typedef __attribute__((ext_vector_type(8))) float v8f;

#define B_  128
#define N_  512
#define M_  512
#define D_  32
#define NTILES ((size_t)B_ * (N_/16) * (M_/16))   // 131072 tiles of 16x16
#define DMAT_BYTES ((size_t)B_ * N_ * M_ * 4)     // 134217728
#define SQ_BYTES   ((size_t)B_ * N_ * 4)          // 262144 each for x,y
#define PART_OFF   (DMAT_BYTES + 2 * SQ_BYTES)
#define WS_NEED    (PART_OFF + B_ * 4)

// ---------------------------------------------------------------------------
// Kernel 0: per-row squared norms for X and Y
// ---------------------------------------------------------------------------
__global__ void sqnorm_kernel(const float* __restrict__ X,
                              const float* __restrict__ Y,
                              float* __restrict__ xsq,
                              float* __restrict__ ysq) {
    int r = blockIdx.x * blockDim.x + threadIdx.x;     // 0 .. 2*B_*N_-1
    int total = B_ * N_;
    const float* src = (r < total) ? (X + (size_t)r * D_)
                                   : (Y + (size_t)(r - total) * D_);
    float s = 0.f;
#pragma unroll
    for (int k = 0; k < D_; k += 4) {
        float4 v = *(const float4*)(src + k);
        s += v.x * v.x + v.y * v.y + v.z * v.z + v.w * v.w;
    }
    if (r < total) xsq[r] = s;
    else           ysq[r - total] = s;
}

// ---------------------------------------------------------------------------
// Kernel 1: D[b,n,m] = xsq[n] + ysq[m] - 2 * <x_n, y_m>  via f32 WMMA
// One wave per 16x16 output tile; 8 chained V_WMMA_F32_16X16X4_F32 (K=32).
// A 16x4 f32 layout: VGPR0 = K0|K2 halves, VGPR1 = K1|K3; M = lane&15.
// B 4x16 f32 layout mirrors it with N = lane&15.
// ---------------------------------------------------------------------------
__global__ void __launch_bounds__(256)
dist_wmma(const float* __restrict__ X, const float* __restrict__ Y,
          const float* __restrict__ xsq, const float* __restrict__ ysq,
          float* __restrict__ Dmat) {
    int wave = threadIdx.x >> 5;
    int lane = threadIdx.x & 31;
    int tile = blockIdx.x * 8 + wave;          // < 131072
    int b  = tile >> 10;                       // 32*32 tiles per batch
    int nt = (tile >> 5) & 31;
    int mt = tile & 31;

    const float* Xb = X + ((size_t)b * N_ + nt * 16) * D_;
    const float* Yb = Y + ((size_t)b * M_ + mt * 16) * D_;

    int row  = lane & 15;
    int koff = (lane >> 4) << 1;               // 0 for lanes 0-15, 2 for 16-31

    v8f acc = {};
#pragma unroll
    for (int k0 = 0; k0 < D_; k0 += 4) {
        v2f a  = *(const v2f*)(Xb + row * D_ + k0 + koff);
        v2f bb = *(const v2f*)(Yb + row * D_ + k0 + koff);
        acc = __builtin_amdgcn_wmma_f32_16x16x4_f32(
            /*neg_a=*/false, a, /*neg_b=*/false, bb,
            /*c_mod=*/(short)0, acc, /*reuse_a=*/false, /*reuse_b=*/false);
    }

    // Epilogue: C/D layout — VGPR i: lanes0-15 -> M=i, lanes16-31 -> M=i+8.
    int mcol = mt * 16 + (lane & 15);
    float ys = ysq[(size_t)b * M_ + mcol];
    int rowbase = nt * 16 + ((lane >> 4) << 3);
    const float* xq = xsq + (size_t)b * N_;
    float* Dp = Dmat + (size_t)b * N_ * M_;
#pragma unroll
    for (int i = 0; i < 8; ++i) {
        int n = rowbase + i;
        Dp[(size_t)n * M_ + mcol] = xq[n] + ys - 2.0f * acc[i];
    }
}

// ---------------------------------------------------------------------------
// Kernel 2: wavefront DP + online logsumexp. One block (512 thr) per batch.
// Thread m owns column m. Step t: thread m computes cell (n=t-m, m).
// 3 rotating anti-diagonal buffers; per-row (mx,sum) online softmin.
// ---------------------------------------------------------------------------
__global__ void __launch_bounds__(512)
dp_kernel(const float* __restrict__ Dmat, float* __restrict__ partial) {
    __shared__ float rbuf[3][M_];
    __shared__ float mx[N_];
    __shared__ float sm[N_];

    const int b = blockIdx.x;
    const int m = threadIdx.x;                 // 0..511
    const float INF = __builtin_inff();

    mx[m] = -INF;
    sm[m] = 0.f;
    __syncthreads();

    const float* Drow = Dmat + (size_t)b * N_ * M_;
    float r_up = INF;                          // R[n-1, m] carried in register
    int w = 0, r1 = 2, r2 = 1;                 // write=t%3, t-1, t-2 buffers

    for (int t = 0; t < N_ + M_ - 1; ++t) {
        int n = t - m;
        if ((unsigned)n < (unsigned)N_) {
            float d    = Drow[n * M_ + m];
            float left = (m == 0) ? INF : rbuf[r1][m - 1];
            float diag;
            if (n == 0) diag = (m == 0) ? 0.f : INF;
            else        diag = (m == 0) ? INF : rbuf[r2][m - 1];
            float up   = (n == 0) ? INF : r_up;

            float r = d + fminf(fminf(diag, up), left);
            rbuf[w][m] = r;
            r_up = r;

            // online logsumexp of -R over row n (rows distinct per thread)
            float v  = -r;
            float cm = mx[n];
            float cs = sm[n];
            if (v > cm) { sm[n] = cs * expf(cm - v) + 1.f; mx[n] = v; }
            else        { sm[n] = cs + expf(v - cm); }
        }
        int tmp = w; w = r2; r2 = r1; r1 = tmp;  // (w,r1,r2) <- (r2,w,r1)
        __syncthreads();
    }

    // softmin[n] = -(mx[n] + log(sm[n])); deterministic tree reduction
    float part = -(mx[m] + logf(sm[m]));
    __syncthreads();
    rbuf[0][m] = part;
    __syncthreads();
    for (int s = 256; s > 0; s >>= 1) {
        if (m < s) rbuf[0][m] += rbuf[0][m + s];
        __syncthreads();
    }
    if (m == 0) partial[b] = rbuf[0][0];
}

// ---------------------------------------------------------------------------
// Fallback fused kernel (ws too small for D): distances on the fly.
// X rows staged in LDS (stride-33 padding: conflict-free diagonal reads),
// each thread keeps its Y row in registers.
// ---------------------------------------------------------------------------
#define XS_STRIDE 33
__global__ void __launch_bounds__(512)
sdtw_fused(const float* __restrict__ X, const float* __restrict__ Y,
           float* __restrict__ partial, float* __restrict__ out_atomic) {
    extern __shared__ float smem[];
    float* Xs   = smem;                                // N_*XS_STRIDE
    float* xq   = Xs + N_ * XS_STRIDE;                 // N_
    float* rb   = xq + N_;                             // 3*M_
    float* mx   = rb + 3 * M_;                         // N_
    float* sm   = mx + N_;                             // N_

    const int b = blockIdx.x;
    const int m = threadIdx.x;
    const float INF = __builtin_inff();

    // stage X row m, compute |x_m|^2; load own Y row into registers
    const float* xrow = X + ((size_t)b * N_ + m) * D_;
    const float* yrow = Y + ((size_t)b * M_ + m) * D_;
    float yreg[D_];
    float xs = 0.f, ysq_s = 0.f;
#pragma unroll
    for (int k = 0; k < D_; ++k) {
        float xv = xrow[k];
        Xs[m * XS_STRIDE + k] = xv;
        xs += xv * xv;
        float yv = yrow[k];
        yreg[k] = yv;
        ysq_s += yv * yv;
    }
    xq[m] = xs;
    mx[m] = -INF;
    sm[m] = 0.f;
    __syncthreads();

    float r_up = INF;
    int w = 0, r1 = 2, r2 = 1;
    for (int t = 0; t < N_ + M_ - 1; ++t) {
        int n = t - m;
        if ((unsigned)n < (unsigned)N_) {
            const float* xr = &Xs[n * XS_STRIDE];
            float dot = 0.f;
#pragma unroll
            for (int k = 0; k < D_; ++k) dot = fmaf(xr[k], yreg[k], dot);
            float d = xq[n] + ysq_s - 2.0f * dot;

            float left = (m == 0) ? INF : rb[r1 * M_ + m - 1];
            float diag;
            if (n == 0) diag = (m == 0) ? 0.f : INF;
            else        diag = (m == 0) ? INF : rb[r2 * M_ + m - 1];
            float up   = (n == 0) ? INF : r_up;

            float r = d + fminf(fminf(diag, up), left);
            rb[w * M_ + m] = r;
            r_up = r;

            float v  = -r;
            float cm = mx[n];
            float cs = sm[n];
            if (v > cm) { sm[n] = cs * expf(cm - v) + 1.f; mx[n] = v; }
            else        { sm[n] = cs + expf(v - cm); }
        }
        int tmp = w; w = r2; r2 = r1; r1 = tmp;
        __syncthreads();
    }

    float part = -(mx[m] + logf(sm[m]));
    __syncthreads();
    rb[m] = part;
    __syncthreads();
    for (int s = 256; s > 0; s >>= 1) {
        if (m < s) rb[m] += rb[m + s];
        __syncthreads();
    }
    if (m == 0) {
        if (partial) partial[b] = rb[0];
        else atomicAdd(out_atomic, rb[0]);
    }
}

__global__ void zero_out(float* o) { *o = 0.f; }

__global__ void finalize(const float* __restrict__ partial, float* __restrict__ out) {
    float s = 0.f;
    for (int i = 0; i < B_; ++i) s += partial[i];   // fixed order: deterministic
    *out = s;
}

// ---------------------------------------------------------------------------
extern "C" void kernel_launch(void* const* d_in, const int* in_sizes, int n_in,
                              void* d_out, int out_size, void* d_ws, size_t ws_size,
                              hipStream_t stream) {
    const float* X = (const float*)d_in[0];
    const float* Y = (const float*)d_in[1];
    float* out = (float*)d_out;

    if (ws_size >= WS_NEED) {
        float* Dmat = (float*)d_ws;
        float* xsq  = (float*)((char*)d_ws + DMAT_BYTES);
        float* ysq  = xsq + B_ * N_;
        float* part = (float*)((char*)d_ws + PART_OFF);

        sqnorm_kernel<<<(2 * B_ * N_) / 256, 256, 0, stream>>>(X, Y, xsq, ysq);
        dist_wmma<<<(int)(NTILES / 8), 256, 0, stream>>>(X, Y, xsq, ysq, Dmat);
        dp_kernel<<<B_, 512, 0, stream>>>(Dmat, part);
        finalize<<<1, 1, 0, stream>>>(part, out);
    } else if (ws_size >= B_ * sizeof(float)) {
        float* part = (float*)d_ws;
        size_t shmem = (N_ * XS_STRIDE + N_ + 3 * M_ + N_ + N_) * sizeof(float);
        sdtw_fused<<<B_, 512, shmem, stream>>>(X, Y, part, nullptr);
        finalize<<<1, 1, 0, stream>>>(part, out);
    } else {
        size_t shmem = (N_ * XS_STRIDE + N_ + 3 * M_ + N_ + N_) * sizeof(float);
        zero_out<<<1, 1, 0, stream>>>(out);
        sdtw_fused<<<B_, 512, shmem, stream>>>(X, Y, nullptr, out);
    }
}